// Polar5GDecoder_24077586662018
// MI455X (gfx1250) — compile-verified
//
#include <hip/hip_runtime.h>
#include <stdint.h>

#define NN       256
#define NHALF    128
#define LISTS    8
#define NSTAGES  8
#define THREADS  128
#define SEGPAD   256            // padded per-list segment pool (live = 255)
#define LLR_MAXF 30.0f
#define PM_LARGE 100000.0f

__device__ __forceinline__ float clipf(float x) {
    return fminf(fmaxf(x, -LLR_MAXF), LLR_MAXF);
}
// log(1 + e^z), numerically stable: max(z,0) + log1p(e^{-|z|})
__device__ __forceinline__ float softplusf(float z) {
    return fmaxf(z, 0.0f) + log1pf(expf(-fabsf(z)));
}
// boxplus: logaddexp(0, x+y) - logaddexp(x, y), inputs clipped (matches _cn_op)
__device__ __forceinline__ float cn_opf(float x, float y) {
    x = clipf(x); y = clipf(y);
    float s   = x + y;
    float la0 = fmaxf(s, 0.0f) + log1pf(expf(-fabsf(s)));
    float lab = fmaxf(x, y)    + log1pf(expf(-fabsf(x - y)));
    return la0 - lab;
}

__global__ __launch_bounds__(THREADS) void polar_scl_kernel(
    const float* __restrict__ llr_in, float* __restrict__ out)
{
    const int b   = blockIdx.x;
    const int tid = threadIdx.x;

    __shared__ __align__(16) float llr_ch[NN];             // channel LLRs (shared)
    __shared__ __align__(16) float seg[2][LISTS][SEGPAD];  // stage segs, ping-pong
    __shared__ __align__(16) unsigned char uh[2][NSTAGES][LISTS][NN];
    __shared__ float pm_sh[LISTS];
    __shared__ float pm_cand[2 * LISTS];
    __shared__ float pm_new[LISTS];
    __shared__ int   sel_src[LISTS];
    __shared__ int   sel_bit[LISTS];
    __shared__ int   best_sh;

    // ---- CDNA5 async global->LDS copy of channel LLRs (ASYNCcnt path) ----
    if (tid < NN / 4) {                                    // 64 lanes x 16B = 1KB
        const float* g = llr_in + (size_t)b * NN + (size_t)tid * 4;
        uint32_t lds = (uint32_t)(uintptr_t)&llr_ch[tid * 4];
        asm volatile("global_load_async_to_lds_b128 %0, %1, off"
                     :: "v"(lds), "v"(g) : "memory");
    }
    asm volatile("s_wait_asynccnt 0x0" ::: "memory");

    // zero both partial-sum ping-pong buffers (b128 stores, static trip count)
    {
        uint4* z = (uint4*)uh;
        #pragma unroll
        for (int r = 0; r < (int)(sizeof(uh) / 16) / THREADS; ++r)
            z[r * THREADS + tid] = make_uint4(0u, 0u, 0u, 0u);
        if (tid < LISTS) pm_sh[tid] = (tid == 0) ? 0.0f : PM_LARGE;
    }
    __syncthreads();

    int cur = 0;

    // f-op (CN): seg_s (or llr_ch at s==8) -> seg_{s-1}, over `le` lists
    auto cn_stage = [&](int s, int le) {
        const int half = 1 << (s - 1);
        const int oin  = (1 << s) - 1;
        const int oout = half - 1;
        for (int t = tid; t < le * half; t += THREADS) {
            int l = t >> (s - 1), j = t & (half - 1);
            float x, y;
            if (s == NSTAGES) { x = llr_ch[j];            y = llr_ch[j + half]; }
            else              { x = seg[cur][l][oin + j]; y = seg[cur][l][oin + half + j]; }
            seg[cur][l][oout + j] = cn_opf(x, y);
        }
        __syncthreads();
    };

    for (int i = 0; i < NN; ++i) {
        // during the frozen prefix all 8 paths are identical: compute 1 list only
        const int le = (i < NHALF) ? 1 : LISTS;

        if (i == 0) {
            for (int s = NSTAGES; s >= 1; --s) cn_stage(s, le);
        } else {
            const int k = __builtin_ctz(i);
            // pending partial-sum combines for just-finished subtrees (inner first)
            for (int s = 1; s <= k; ++s) {
                const int half = 1 << (s - 1);
                const int lo   = i - (1 << s);
                for (int t = tid; t < le * half; t += THREADS) {
                    int l = t >> (s - 1), j = t & (half - 1);
                    unsigned char ul = uh[cur][s - 1][l][lo + j];
                    unsigned char ur = uh[cur][s - 1][l][lo + half + j];
                    uh[cur][s][l][lo + j]        = (unsigned char)(ul ^ ur);
                    uh[cur][s][l][lo + half + j] = ur;
                }
                __syncthreads();
            }
            // g-op (VN) at stage k+1 into the right child's segment
            {
                const int s    = k + 1;
                const int half = 1 << k;
                const int lo   = i - half;            // left-half base of the node
                const int oin  = (1 << s) - 1;
                const int oout = half - 1;
                for (int t = tid; t < le * half; t += THREADS) {
                    int l = t >> k, j = t & (half - 1);
                    float x, y;
                    if (s == NSTAGES) { x = llr_ch[j];            y = llr_ch[j + half]; }
                    else              { x = seg[cur][l][oin + j]; y = seg[cur][l][oin + half + j]; }
                    float u = (float)uh[cur][s - 1][l][lo + j];
                    seg[cur][l][oout + j] = fmaf(1.0f - 2.0f * u, x, y);
                }
                __syncthreads();
            }
            for (int s = k; s >= 1; --s) cn_stage(s, le);
        }

        // ---------------- leaf i ----------------
        if (i < NHALF) {
            // frozen: bit stays 0; paths identical, so one leaf value serves all
            if (tid < LISTS) {
                float L = clipf(seg[cur][0][0]);
                pm_sh[tid] += softplusf(-L);
            }
            __syncthreads();
        } else {
            const int nxt = cur ^ 1;
            // 16 candidate metrics: c = bit*8 + src  (matches concat([pm0, pm1]))
            if (tid < 2 * LISTS) {
                int srcl = tid & 7, bit = tid >> 3;
                float L = clipf(seg[cur][srcl][0]);
                pm_cand[tid] = pm_sh[srcl] + softplusf(bit ? L : -L);
            }
            __syncthreads();
            // stable ascending rank (mimics jnp.argsort), keep the 8 smallest
            if (tid < 2 * LISTS) {
                float me = pm_cand[tid];
                int rank = 0;
                for (int j = 0; j < 2 * LISTS; ++j) {
                    float pj = pm_cand[j];
                    rank += (pj < me) || (pj == me && j < tid);
                }
                if (rank < LISTS) {
                    sel_src[rank] = tid & 7;
                    sel_bit[rank] = tid >> 3;
                    pm_new[rank]  = me;
                }
            }
            __syncthreads();
            // clone surviving paths: b128 gather of LLR segments across lists
            {
                const uint4* s128 = (const uint4*)&seg[cur][0][0];
                uint4*       d128 = (uint4*)&seg[nxt][0][0];
                for (int t = tid; t < LISTS * (SEGPAD / 4); t += THREADS) {
                    int q = t & 63, l = t >> 6;               // 64 x b128 per list
                    d128[t] = s128[(sel_src[l] << 6) + q];
                }
            }
            // clone partial-sum planes: b128 gather (16 x b128 per (stage,list))
            {
                const uint4* su = (const uint4*)&uh[cur][0][0][0];
                uint4*       du = (uint4*)&uh[nxt][0][0][0];
                for (int t = tid; t < NSTAGES * LISTS * (NN / 16); t += THREADS) {
                    int q  = t & 15;
                    int l  = (t >> 4) & (LISTS - 1);
                    int st = t >> 7;
                    du[t] = su[((st << 3) + sel_src[l]) * 16 + q];
                }
            }
            __syncthreads();
            if (tid < LISTS) {
                pm_sh[tid] = pm_new[tid];
                uh[nxt][0][tid][i] = (unsigned char)sel_bit[tid];
            }
            __syncthreads();
            cur = nxt;
        }
    }

    // best path = first argmin of pm (stable, like jnp.argmin)
    if (tid == 0) {
        int best = 0; float bp = pm_sh[0];
        for (int l = 1; l < LISTS; ++l) if (pm_sh[l] < bp) { bp = pm_sh[l]; best = l; }
        best_sh = best;
    }
    __syncthreads();
    const int best = best_sh;
    for (int t = tid; t < NHALF; t += THREADS)
        out[(size_t)b * NHALF + t] = (float)uh[cur][0][best][NHALF + t];
}

extern "C" void kernel_launch(void* const* d_in, const int* in_sizes, int n_in,
                              void* d_out, int out_size, void* d_ws, size_t ws_size,
                              hipStream_t stream) {
    const float* llr = (const float*)d_in[0];
    float*       o   = (float*)d_out;
    const int B = in_sizes[0] / NN;   // 512 codewords
    polar_scl_kernel<<<B, THREADS, 0, stream>>>(llr, o);
}